// OrderingNetV4_75849122447995
// MI455X (gfx1250) — compile-verified
//
#include <hip/hip_runtime.h>
#include <hip/hip_bf16.h>
#include <math.h>

typedef float v2f __attribute__((ext_vector_type(2)));
typedef float v8f __attribute__((ext_vector_type(8)));

#define GG 512   // G
#define CC 384   // C
#define HH 768   // H
#define BB 32    // batch
#define MTOT (BB * GG)  // 16384 rows

// ---------------------------------------------------------------------------
// FP32 WMMA GEMM: D = epilogue(A @ Bm + bias)
//   A: MxK row-major, Bm: KxN row-major, D: MxN row-major
//   Each wave computes a 64x64 tile via 4x4 subtiles of V_WMMA_F32_16X16X4_F32.
//   Block = 128 threads (4 waves) -> 64 rows x 256 cols per block.
//   mode 0: relu(x + bias)    mode 1: (x + bias) * invTau
// ---------------------------------------------------------------------------
__global__ __launch_bounds__(128)
void wmma_gemm_f32(const float* __restrict__ A, const float* __restrict__ Bm,
                   const float* __restrict__ bias, float* __restrict__ D,
                   int M, int N, int K, int mode, float invTau) {
  const int lane = threadIdx.x & 31;
  const int wave = threadIdx.x >> 5;
  const int row0 = blockIdx.x * 64;
  const int col0 = blockIdx.y * 256 + wave * 64;
  if (col0 >= N || row0 >= M) return;

  v8f c[4][4] = {};

  // A frag (16x4 fp32): lanes 0-15 -> M=lane, K pair {0,1}; lanes 16-31 -> K pair {2,3}
  const int mA = lane & 15;
  const int kA = (lane >> 4) * 2;
  // B frag (4x16 fp32): lanes 0-15 -> N=lane, K pair {0,1}; lanes 16-31 -> K pair {2,3}
  const int nB = lane & 15;
  const int kB = (lane >> 4) * 2;

  for (int k = 0; k < K; k += 4) {
    v2f a[4], b[4];
#pragma unroll
    for (int i = 0; i < 4; ++i) {
      const float* ap = A + (size_t)(row0 + i * 16 + mA) * K + k + kA;
      a[i].x = ap[0];
      a[i].y = ap[1];
    }
#pragma unroll
    for (int j = 0; j < 4; ++j) {
      const float* bp = Bm + (size_t)(k + kB) * N + (col0 + j * 16 + nB);
      b[j].x = bp[0];
      b[j].y = bp[N];  // next K row
    }
#pragma unroll
    for (int i = 0; i < 4; ++i)
#pragma unroll
      for (int j = 0; j < 4; ++j)
        c[i][j] = __builtin_amdgcn_wmma_f32_16x16x4_f32(
            /*neg_a=*/false, a[i], /*neg_b=*/false, b[j],
            /*c_mod=*/(short)0, c[i][j], /*reuse_a=*/false, /*reuse_b=*/false);
  }

  // C/D layout: VGPR r, lane<16 -> (M=r, N=lane); lane>=16 -> (M=8+r, N=lane-16)
  const int mOff = (lane >> 4) * 8;
  const int nOff = lane & 15;
#pragma unroll
  for (int j = 0; j < 4; ++j) {
    const int col = col0 + j * 16 + nOff;
    const float bv = bias[col];
#pragma unroll
    for (int i = 0; i < 4; ++i) {
#pragma unroll
      for (int r = 0; r < 8; ++r) {
        float x = c[i][j][r] + bv;
        x = (mode == 0) ? fmaxf(x, 0.0f) : x * invTau;
        D[(size_t)(row0 + i * 16 + mOff + r) * N + col] = x;
      }
    }
  }
}

// ---------------------------------------------------------------------------
// Sinkhorn row pass: log_alpha[row] -= logsumexp(log_alpha[row]).
// One wave per row (G=512 -> 16 elements/lane, kept in registers).
// ---------------------------------------------------------------------------
__global__ __launch_bounds__(256)
void row_lse(float* __restrict__ LA, int rows) {
  const int w = blockIdx.x * (blockDim.x >> 5) + (threadIdx.x >> 5);
  if (w >= rows) return;
  const int lane = threadIdx.x & 31;
  float* row = LA + (size_t)w * GG;

  float v[16];
  float mx = -INFINITY;
#pragma unroll
  for (int i = 0; i < 16; ++i) {
    v[i] = row[lane + i * 32];
    mx = fmaxf(mx, v[i]);
  }
  for (int off = 16; off; off >>= 1) mx = fmaxf(mx, __shfl_xor(mx, off, 32));
  float s = 0.0f;
#pragma unroll
  for (int i = 0; i < 16; ++i) s += expf(v[i] - mx);
  for (int off = 16; off; off >>= 1) s += __shfl_xor(s, off, 32);
  const float lse = mx + logf(s);
#pragma unroll
  for (int i = 0; i < 16; ++i) row[lane + i * 32] = v[i] - lse;
}

// ---------------------------------------------------------------------------
// Sinkhorn column pass: one thread per (batch, col), two-pass max/sum then
// subtract. Adjacent threads hit adjacent columns -> coalesced strided loads.
// ---------------------------------------------------------------------------
__global__ __launch_bounds__(256)
void col_lse(float* __restrict__ LA) {
  const int col = blockIdx.x * blockDim.x + threadIdx.x;
  float* base = LA + (size_t)blockIdx.y * GG * GG + col;
  float mx = -INFINITY;
  for (int i = 0; i < GG; ++i) mx = fmaxf(mx, base[(size_t)i * GG]);
  float s = 0.0f;
  for (int i = 0; i < GG; ++i) s += expf(base[(size_t)i * GG] - mx);
  const float lse = mx + logf(s);
  for (int i = 0; i < GG; ++i) base[(size_t)i * GG] -= lse;
}

// ---------------------------------------------------------------------------
// Greedy assignment (per batch): repeatedly take the global argmax over
// (free row, free col). exp() is skipped (monotone under argmax) -> runs on
// log_alpha directly.
//
// Register-resident cached bests: wave w owns rows [32w, 32w+32), lane l
// holds (bestVal, bestCol) of row 32w+l in registers. Per step:
//   1) each wave: 5-level shfl_xor argmax over its 32 registers (no barrier),
//      lane 0 posts one partial (val,row,col) to LDS
//   2) one barrier; wave 0 shfl-reduces the 16 partials, publishes pick,
//      retires the column
//   3) one barrier; owning lane kills the picked row; rows whose cached best
//      column was taken are rescanned cooperatively by their wave (ballot
//      loop), the xor-butterfly leaves the result on every lane so the owning
//      lane just captures it.
// Only 2 barriers per step (vs ~10 for an LDS tree). Ties break toward
// smaller row then smaller column (matches jnp.argmax flat-index order).
// ---------------------------------------------------------------------------
__global__ __launch_bounds__(512)
void greedy_assign(const float* __restrict__ P, int* __restrict__ perm) {
  __shared__ unsigned char sColFree[GG];
  __shared__ float sPartVal[16];
  __shared__ int   sPartRow[16];
  __shared__ int   sPartCol[16];
  __shared__ int   sPickRow, sPickCol;

  const int b    = blockIdx.x;
  const int tid  = threadIdx.x;
  const int lane = tid & 31;
  const int wv   = tid >> 5;           // 0..15
  const int r0   = wv * 32;            // first row owned by this wave
  const float* Pb = P + (size_t)b * GG * GG;

  sColFree[tid] = 1;
  __syncthreads();

  // Init: cooperative per-row argmax over all 512 columns; result lands on
  // every lane of the butterfly, owning lane captures its row's best.
  float myBestVal = -INFINITY;
  int   myBestCol = 0;
  for (int k = 0; k < 32; ++k) {
    const float* rowp = Pb + (size_t)(r0 + k) * GG;
    float bvv = -INFINITY;
    int bc = 0;
#pragma unroll
    for (int i = 0; i < 16; ++i) {
      const int cc = lane + i * 32;
      const float vv = rowp[cc];
      if (vv > bvv) { bvv = vv; bc = cc; }
    }
#pragma unroll
    for (int off = 16; off; off >>= 1) {
      const float ov = __shfl_xor(bvv, off, 32);
      const int   oc = __shfl_xor(bc, off, 32);
      if (ov > bvv || (ov == bvv && oc < bc)) { bvv = ov; bc = oc; }
    }
    if (lane == k) { myBestVal = bvv; myBestCol = bc; }
  }
  __syncthreads();

  for (int t = 0; t < GG; ++t) {
    // (1) wave-local argmax over the 32 register-resident row bests.
    {
      float bvv = myBestVal;
      int   br  = r0 + lane;
      int   bc  = myBestCol;
#pragma unroll
      for (int off = 16; off; off >>= 1) {
        const float ov = __shfl_xor(bvv, off, 32);
        const int   orr = __shfl_xor(br, off, 32);
        const int   oc = __shfl_xor(bc, off, 32);
        if (ov > bvv || (ov == bvv && orr < br)) { bvv = ov; br = orr; bc = oc; }
      }
      if (lane == 0) { sPartVal[wv] = bvv; sPartRow[wv] = br; sPartCol[wv] = bc; }
    }
    __syncthreads();

    // (2) wave 0 reduces the 16 partials and publishes the pick.
    if (wv == 0) {
      float bvv = (lane < 16) ? sPartVal[lane] : -INFINITY;
      int   br  = (lane < 16) ? sPartRow[lane] : 0x7FFFFFFF;
      int   bc  = (lane < 16) ? sPartCol[lane] : 0;
#pragma unroll
      for (int off = 16; off; off >>= 1) {
        const float ov = __shfl_xor(bvv, off, 32);
        const int   orr = __shfl_xor(br, off, 32);
        const int   oc = __shfl_xor(bc, off, 32);
        if (ov > bvv || (ov == bvv && orr < br)) { bvv = ov; br = orr; bc = oc; }
      }
      if (lane == 0) {
        sPickRow = br;
        sPickCol = bc;
        perm[b * GG + br] = bc;
        sColFree[bc] = 0;
      }
    }
    __syncthreads();

    // (3) retire the picked row; rescan rows whose cached best col was taken.
    const int pr = sPickRow;
    const int pc = sPickCol;
    if (r0 + lane == pr) myBestVal = -INFINITY;
    const bool need = (myBestVal != -INFINITY) && (myBestCol == pc);
    unsigned long long ball = __ballot(need);
    while (ball) {
      const int k = __ffsll(ball) - 1;
      ball &= ball - 1;
      const float* rowp = Pb + (size_t)(r0 + k) * GG;
      float bvv = -INFINITY;
      int bc = -1;
#pragma unroll
      for (int i = 0; i < 16; ++i) {
        const int cc = lane + i * 32;
        const float vv = sColFree[cc] ? rowp[cc] : -INFINITY;
        if (vv > bvv) { bvv = vv; bc = cc; }
      }
#pragma unroll
      for (int off = 16; off; off >>= 1) {
        const float ov = __shfl_xor(bvv, off, 32);
        const int   oc = __shfl_xor(bc, off, 32);
        if (ov > bvv || (ov == bvv && oc < bc)) { bvv = ov; bc = oc; }
      }
      if (lane == k) { myBestVal = bvv; myBestCol = bc; }
    }
    // No trailing barrier needed: step (1) touches only this wave's registers,
    // and the next write to shared state (wave 0 in step (2)) is fenced by the
    // barrier after step (1).
  }
}

// ---------------------------------------------------------------------------
// Apply permutation: out[perm[r]] = in[r]; also emit perm as float.
// Grid (G, B), block 128.
// ---------------------------------------------------------------------------
__global__ __launch_bounds__(128)
void scatter_out(const float* __restrict__ coords, const float* __restrict__ feats,
                 const int* __restrict__ perm, float* __restrict__ outCoords,
                 float* __restrict__ outFeats, float* __restrict__ outPerm) {
  const int b = blockIdx.y;
  const int r = blockIdx.x;
  const int tid = threadIdx.x;
  const int p = perm[b * GG + r];
  if (tid < 3) outCoords[((size_t)b * GG + p) * 3 + tid] = coords[((size_t)b * GG + r) * 3 + tid];
  if (tid == 3) outPerm[b * GG + r] = (float)p;
  for (int i = tid; i < CC; i += 128)
    outFeats[((size_t)b * GG + p) * CC + i] = feats[((size_t)b * GG + r) * CC + i];
}

extern "C" void kernel_launch(void* const* d_in, const int* in_sizes, int n_in,
                              void* d_out, int out_size, void* d_ws, size_t ws_size,
                              hipStream_t stream) {
  (void)in_sizes; (void)n_in; (void)out_size; (void)ws_size;

  const float* coords = (const float*)d_in[0];  // (B,G,3)
  const float* feats  = (const float*)d_in[1];  // (B,G,C)
  const float* W1     = (const float*)d_in[2];  // (C,H)
  const float* b1     = (const float*)d_in[3];  // (H,)
  const float* W2     = (const float*)d_in[4];  // (H,G)
  const float* b2     = (const float*)d_in[5];  // (G,)

  float* out       = (float*)d_out;
  float* outCoords = out;                                   // B*G*3
  float* outFeats  = out + (size_t)BB * GG * 3;             // B*G*C
  float* outPerm   = outFeats + (size_t)BB * GG * CC;       // B*G

  char* ws = (char*)d_ws;
  float* h  = (float*)ws;                                       // MTOT*HH f32 (50.3 MB)
  float* la = (float*)(ws + (size_t)MTOT * HH * 4);             // MTOT*GG f32 (33.5 MB)
  int* perm = (int*)(ws + (size_t)MTOT * HH * 4 + (size_t)MTOT * GG * 4);

  // GEMM1: h = relu(feats @ W1 + b1)
  {
    dim3 grid(MTOT / 64, HH / 256);  // 256 x 3
    wmma_gemm_f32<<<grid, 128, 0, stream>>>(feats, W1, b1, h, MTOT, HH, CC, 0, 1.0f);
  }
  // GEMM2: la = (h @ W2 + b2) / TAU   (log_alpha seed)
  {
    dim3 grid(MTOT / 64, GG / 256);  // 256 x 2
    wmma_gemm_f32<<<grid, 128, 0, stream>>>(h, W2, b2, la, MTOT, GG, HH, 1, 10.0f);
  }
  // Sinkhorn: 10 x (row LSE, col LSE). exp() is skipped (monotone under argmax).
  for (int it = 0; it < 10; ++it) {
    row_lse<<<MTOT / 8, 256, 0, stream>>>(la, MTOT);
    col_lse<<<dim3(GG / 256, BB), 256, 0, stream>>>(la);
  }
  // Greedy assignment per batch.
  greedy_assign<<<BB, GG, 0, stream>>>(la, perm);
  // Scatter outputs (full permutation -> every output element written).
  scatter_out<<<dim3(GG, BB), 128, 0, stream>>>(coords, feats, perm,
                                                outCoords, outFeats, outPerm);
}